// GraphAttentionLayer_77068893160074
// MI455X (gfx1250) — compile-verified
//
#include <hip/hip_runtime.h>

// ============================================================================
// GAT layer on MI455X (gfx1250).
//
// Key simplification (mathematically exact, not an approximation):
//   softmax over a singleton axis == 1.0, so alpha == 1 for every edge and
//   a_w / a_b / leaky_relu drop out of the result entirely. The layer is:
//       h   = x @ W_w + W_b                       (fp32 GEMM, via f32 WMMA)
//       out = segment_sum(h[col], row)            (gather + atomic scatter-add)
//
// Roofline: GEMM is 1.64 GFLOP (negligible). Edge phase: ~410 MB gathers +
// ~102M f32 atomics, all resident in the 192 MB L2 -> L2/atomic bound.
// ============================================================================

typedef __attribute__((ext_vector_type(2))) float v2f;
typedef __attribute__((ext_vector_type(8))) float v8f;

#define GAT_F_IN  128
#define GAT_F_OUT 64

// ----------------------------------------------------------------------------
// h = x @ W + bias using V_WMMA_F32_16X16X4_F32 (wave32).
// Block = 128 threads = 4 waves; wave w computes the 16x16 tile at columns
// n0 = 16*w (covering F_OUT=64), rows m0..m0+15 where m0 = 16*blockIdx.x.
//
// f32 WMMA operand layouts (CDNA5 ISA 7.12.2):
//   A (16x4, M x K): lanes 0-15 hold M=lane, {K0,K1} in v[0],v[1];
//                    lanes 16-31 hold M=lane-16, {K2,K3}.
//   B (4x16, K x N): lanes 0-15 hold N=lane, rows {K0,K1} in v[0],v[1];
//                    lanes 16-31 hold N=lane-16, rows {K2,K3}.
//   C/D (16x16):     VGPR i holds M = i + 8*(lane>=16), N = lane&15.
// ----------------------------------------------------------------------------
__global__ __launch_bounds__(128) void gat_gemm_wmma(
    const float* __restrict__ x,      // [n_nodes, 128]
    const float* __restrict__ W,      // [128, 64] row-major
    const float* __restrict__ bias,   // [64]
    float* __restrict__ h,            // [n_nodes, 64]
    int n_nodes)
{
    const int lane  = threadIdx.x & 31;
    const int wave  = threadIdx.x >> 5;
    const int n0    = wave * 16;          // output-column tile (F_OUT/16 == 4 waves)
    const int m0    = blockIdx.x * 16;    // node-row tile

    const int mlane = lane & 15;
    const int hsel  = lane >> 4;          // 0 -> K pair {0,1}, 1 -> K pair {2,3}

    int mrow = m0 + mlane;
    if (mrow >= n_nodes) mrow = n_nodes - 1;   // clamp loads (stores are guarded)

    const float* xrow = x + (long)mrow * GAT_F_IN;   // contiguous in K -> b64 loads
    const float* wcol = W + n0 + mlane;              // column n, stride F_OUT per K

    v8f acc = {};
#pragma unroll
    for (int kk = 0; kk < GAT_F_IN / 4; ++kk) {
        const int k0 = kk * 4 + hsel * 2;
        v2f a = *(const v2f*)(xrow + k0);            // A[m][k0], A[m][k0+1]
        v2f b;
        b.x = wcol[(long)(k0    ) * GAT_F_OUT];      // B[k0  ][n]
        b.y = wcol[(long)(k0 + 1) * GAT_F_OUT];      // B[k0+1][n]
        // (neg_a, A, neg_b, B, c_mod, C, reuse_a, reuse_b)
        acc = __builtin_amdgcn_wmma_f32_16x16x4_f32(
            false, a, false, b, (short)0, acc, false, false);
    }

    const float bn = bias[n0 + mlane];
    float* hp = h + (long)m0 * GAT_F_OUT + n0 + mlane;
#pragma unroll
    for (int i = 0; i < 8; ++i) {
        const int mrel = i + hsel * 8;               // D-matrix row for VGPR i
        if (m0 + mrel < n_nodes)
            hp[(long)mrel * GAT_F_OUT] = acc[i] + bn;
    }
}

// ----------------------------------------------------------------------------
// out[row[e]] += h[col[e]]   (alpha == 1 for every edge, see header comment)
// 16 threads per edge, one float4 each: coalesced 256B gather of h[col],
// 4 contiguous global_atomic_add_f32 into out[row] (wave covers cachelines).
// ----------------------------------------------------------------------------
__global__ __launch_bounds__(256) void gat_scatter(
    const float* __restrict__ h,              // [n_nodes, 64]
    const long long* __restrict__ edge_index, // [2, n_edges] int64
    float* __restrict__ out,                  // [n_nodes, 64]
    int n_edges)
{
    const long gid  = (long)blockIdx.x * blockDim.x + threadIdx.x;
    const int  e    = (int)(gid >> 4);
    const int  part = (int)(gid & 15);
    if (e >= n_edges) return;

    const int r = (int)edge_index[e];            // scatter destination node
    const int c = (int)edge_index[n_edges + e];  // feature source node

    const float4 v = *(const float4*)(h + (long)c * GAT_F_OUT + part * 4);
    float* o = out + (long)r * GAT_F_OUT + part * 4;
    atomicAdd(o + 0, v.x);
    atomicAdd(o + 1, v.y);
    atomicAdd(o + 2, v.z);
    atomicAdd(o + 3, v.w);
}

// ----------------------------------------------------------------------------
// d_out is poisoned (0xAA) by the harness; zero it before the scatter-add.
// ----------------------------------------------------------------------------
__global__ __launch_bounds__(256) void gat_zero(float4* __restrict__ out, int n4)
{
    const long gid = (long)blockIdx.x * blockDim.x + threadIdx.x;
    if (gid < n4) out[gid] = make_float4(0.f, 0.f, 0.f, 0.f);
}

extern "C" void kernel_launch(void* const* d_in, const int* in_sizes, int n_in,
                              void* d_out, int out_size, void* d_ws, size_t ws_size,
                              hipStream_t stream) {
    // setup_inputs() order: x, edge_index, W_w, W_b, a_w, a_b
    const float*     x      = (const float*)d_in[0];
    const long long* eidx   = (const long long*)d_in[1];   // int64 [2, E]
    const float*     W_w    = (const float*)d_in[2];       // [128, 64]
    const float*     W_b    = (const float*)d_in[3];       // [64]
    // d_in[4] (a_w) and d_in[5] (a_b) are mathematically dead: softmax over a
    // size-1 axis makes every attention weight exactly 1.0.

    const int n_nodes = in_sizes[0] / GAT_F_IN;
    const int n_edges = in_sizes[1] / 2;

    float* out = (float*)d_out;
    float* h   = (float*)d_ws;   // [n_nodes, 64] fp32 scratch (25.6 MB)

    // 1) zero the (poisoned) output accumulator
    {
        const int n4 = out_size / 4;
        const int grid = (n4 + 255) / 256;
        gat_zero<<<grid, 256, 0, stream>>>((float4*)out, n4);
    }

    // 2) h = x @ W + b  (WMMA f32)
    {
        const int grid = (n_nodes + 15) / 16;
        gat_gemm_wmma<<<grid, 128, 0, stream>>>(x, W_w, W_b, h, n_nodes);
    }

    // 3) out[row] += h[col]  over all edges
    {
        const long total = (long)n_edges * 16;
        const int grid = (int)((total + 255) / 256);
        gat_scatter<<<grid, 256, 0, stream>>>(h, eidx, out, n_edges);
    }
}